// AngularPenaltySMLoss_6193342841668
// MI455X (gfx1250) — compile-verified
//
#include <hip/hip_runtime.h>
#include <hip/hip_bf16.h>
#include <stdint.h>

#define THREADS   256
#define CHUNK_F4  1024                 // float4 per chunk = 4096 floats = 16 KB
#define PER_THR   (CHUNK_F4 / THREADS) // 4 async b128 loads per thread per chunk
#define NBUF      4                    // 4 LDS buffers -> 3 chunks in flight

__device__ __forceinline__ float exp_s(float x) {
    // exp(20*x) = exp2(20*log2(e)*x) -> single v_exp_f32
    return __builtin_amdgcn_exp2f(x * 28.853900817779268f);
}

__global__ __launch_bounds__(THREADS)
void cosface_rowloss_kernel(const float* __restrict__ wf,
                            const int* __restrict__ labels,
                            float* __restrict__ row_loss,
                            int B, int C) {
    __shared__ float4 lbuf[NBUF][CHUNK_F4];       // 64 KB quad buffer
    __shared__ float  sred[THREADS / 32];

    const int b   = blockIdx.x;
    const int tid = threadIdx.x;
    const float*   row       = wf + (size_t)b * (size_t)C;
    const uint64_t rowbase   = (uint64_t)(uintptr_t)row;
    const uint32_t row_bytes = (uint32_t)C * 4u;
    const int      nchunk    = (C + CHUNK_F4 * 4 - 1) / (CHUNK_F4 * 4);
    const int      nfull     = C / (CHUNK_F4 * 4); // chunks needing no bounds checks

    // Issue one chunk: exactly PER_THR async b128 loads per thread (uniform
    // per-wave ASYNCcnt increment of 4). GVS addressing: uniform 64-bit SGPR
    // row base + per-lane 32-bit byte offset. Out-of-row offsets clamped
    // in-bounds (duplicated values masked out at accumulate time).
    auto issue_chunk = [&](int k, int buf) {
        const uint32_t cb = (uint32_t)k * (CHUNK_F4 * 16u);
        #pragma unroll
        for (int i = 0; i < PER_THR; ++i) {
            const int      j    = tid + i * THREADS;
            uint32_t       goff = cb + (uint32_t)j * 16u;
            if (goff > row_bytes - 16u) goff = row_bytes - 16u;
            const uint32_t laddr = (uint32_t)(uintptr_t)&lbuf[buf][j];
            asm volatile("global_load_async_to_lds_b128 %0, %1, %2"
                         :
                         : "v"(laddr), "v"(goff), "s"(rowbase)
                         : "memory");
        }
    };

    // Prime the pipeline: up to 3 chunks in flight (ASYNCcnt = 12 per wave).
    #pragma unroll
    for (int p = 0; p < NBUF - 1; ++p)
        if (p < nchunk) issue_chunk(p, p);

    float4 acc4 = make_float4(0.0f, 0.0f, 0.0f, 0.0f);

    for (int k = 0; k < nchunk; ++k) {
        // Async loads complete in order; oldest chunk (k) is done when the
        // counter drops to 4 * (#chunks still in flight behind it).
        const int rem = nchunk - 1 - k;
        if (rem >= 2) {
            asm volatile("s_wait_asynccnt 0x8" ::: "memory");
        } else if (rem == 1) {
            asm volatile("s_wait_asynccnt 0x4" ::: "memory");
        } else {
            asm volatile("s_wait_asynccnt 0x0" ::: "memory");
        }

        // Refill early: target buffer (k+3)&3 == (k-1)&3 was fully consumed
        // in the previous iteration ("memory" clobbers keep the ordering).
        if (k + (NBUF - 1) < nchunk) issue_chunk(k + (NBUF - 1), (k + (NBUF - 1)) & (NBUF - 1));

        const int buf = k & (NBUF - 1);
        if (k < nfull) {
            // Fast path: full chunk, no bounds checks.
            #pragma unroll
            for (int i = 0; i < PER_THR; ++i) {
                const int    j = tid + i * THREADS;
                const float4 v = lbuf[buf][j];              // ds_load_b128
                acc4.x += exp_s(v.x);
                acc4.y += exp_s(v.y);
                acc4.z += exp_s(v.z);
                acc4.w += exp_s(v.w);
            }
        } else {
            // Tail chunk: per-element column masking.
            #pragma unroll
            for (int i = 0; i < PER_THR; ++i) {
                const int    j  = tid + i * THREADS;
                const float4 v  = lbuf[buf][j];
                const int    c0 = k * (CHUNK_F4 * 4) + j * 4;
                if (c0 + 0 < C) acc4.x += exp_s(v.x);
                if (c0 + 1 < C) acc4.y += exp_s(v.y);
                if (c0 + 2 < C) acc4.z += exp_s(v.z);
                if (c0 + 3 < C) acc4.w += exp_s(v.w);
            }
        }
    }

    float acc = (acc4.x + acc4.y) + (acc4.z + acc4.w);

    // wave32 shuffle reduction, then cross-wave via LDS
    #pragma unroll
    for (int off = 16; off > 0; off >>= 1)
        acc += __shfl_down(acc, off, 32);
    const int wid  = tid >> 5;
    const int lane = tid & 31;
    if (lane == 0) sred[wid] = acc;
    __syncthreads();

    if (tid == 0) {
        float rowsum = 0.0f;
        #pragma unroll
        for (int w = 0; w < THREADS / 32; ++w) rowsum += sred[w];
        const int   lab    = labels[b];
        const float target = row[lab];
        const float numer  = 20.0f * (target - 0.0f);       // S*(t - M), M = 0
        const float excl   = rowsum - exp_s(target);        // row_sum - exp(S*t)
        const float denom  = __builtin_amdgcn_exp2f(numer * 1.4426950408889634f) + excl;
        const float lnd    = __builtin_amdgcn_logf(denom) * 0.6931471805599453f;
        row_loss[b] = numer - lnd;
    }
}

__global__ __launch_bounds__(THREADS)
void reduce_neg_mean_kernel(const float* __restrict__ row_loss,
                            float* __restrict__ out, int B) {
    __shared__ float sred[THREADS / 32];
    float acc = 0.0f;
    for (int i = threadIdx.x; i < B; i += THREADS) acc += row_loss[i];
    #pragma unroll
    for (int off = 16; off > 0; off >>= 1)
        acc += __shfl_down(acc, off, 32);
    const int wid  = threadIdx.x >> 5;
    const int lane = threadIdx.x & 31;
    if (lane == 0) sred[wid] = acc;
    __syncthreads();
    if (threadIdx.x == 0) {
        float total = 0.0f;
        #pragma unroll
        for (int w = 0; w < THREADS / 32; ++w) total += sred[w];
        out[0] = -total / (float)B;
    }
}

extern "C" void kernel_launch(void* const* d_in, const int* in_sizes, int n_in,
                              void* d_out, int out_size, void* d_ws, size_t ws_size,
                              hipStream_t stream) {
    const float* wf     = (const float*)d_in[0];
    const int*   labels = (const int*)d_in[1];   // JAX default x64-off -> int32
    const int B = in_sizes[1];
    const int C = in_sizes[0] / B;

    float* row_loss = (float*)d_ws;              // B floats = 32 KB scratch

    cosface_rowloss_kernel<<<B, THREADS, 0, stream>>>(wf, labels, row_loss, B, C);
    reduce_neg_mean_kernel<<<1, THREADS, 0, stream>>>(row_loss, (float*)d_out, B);
}